// LeanPAMLayer_33079838114541
// MI455X (gfx1250) — compile-verified
//
#include <hip/hip_runtime.h>
#include <hip/hip_bf16.h>
#include <math.h>

typedef __attribute__((ext_vector_type(2))) float v2f;
typedef __attribute__((ext_vector_type(4))) float v4f;
typedef __attribute__((ext_vector_type(8))) float v8f;

#define B_     4
#define T_     2048
#define DIM_   1024
#define H_     16
#define HD_    64
#define INNER_ 1024
#define CK     128            // kernel-internal chunk (mathematically equivalent rechunking)
#define NCHUNK (T_ / CK)
#define NREGQ  ((3 * INNER_) / 64)   // 48 N-regions of 64 cols for qkv
#define NREGO  (DIM_ / 64)           // 16 N-regions for out-proj

#if __has_builtin(__builtin_amdgcn_global_load_async_to_lds_b128)
#define ASYNC_COPY 1
// Parameter type per hipcc diagnostic: pointer to 16B int-vector granule.
typedef __attribute__((__vector_size__(4 * sizeof(int)))) int ai4;
typedef __attribute__((address_space(3))) ai4* lds_ai4_p;
#endif

// fp32 WMMA: D(16x16,f32) = A(16x4,f32) x B(4x16,f32) + C
__device__ __forceinline__ v8f wmma4(v2f a, v2f b, v8f c) {
  return __builtin_amdgcn_wmma_f32_16x16x4_f32(false, a, false, b, (short)0, c,
                                               false, false);
}

// ---------------------------------------------------------------------------
// Gate: dt = softplus([xr,xi] @ w_dt.T + b_dt + dt_bias), p = sigmoid(|x| @ w_pg.T + b_pg)
// gamma = exp(-dt)*(1-p)+p.  One thread per (b,t); weights broadcast across lanes.
// ---------------------------------------------------------------------------
__global__ __launch_bounds__(256) void gate_kernel(
    const float* __restrict__ x, const float* __restrict__ w_dt,
    const float* __restrict__ b_dt, const float* __restrict__ dt_bias,
    const float* __restrict__ w_pg, const float* __restrict__ b_pg,
    float* __restrict__ p_out, float* __restrict__ g_out) {
  int bt = blockIdx.x * blockDim.x + threadIdx.x;
  if (bt >= B_ * T_) return;
  const float* xp = x + (size_t)bt * DIM_ * 2;
  float adt[H_], ap[H_];
#pragma unroll
  for (int h = 0; h < H_; ++h) { adt[h] = 0.f; ap[h] = 0.f; }
  for (int k = 0; k < DIM_; ++k) {
    float xr = xp[2 * k], xi = xp[2 * k + 1];
    float mag = sqrtf(xr * xr + xi * xi + 1e-8f);
#pragma unroll
    for (int h = 0; h < H_; ++h) {
      adt[h] += xr * w_dt[h * 2 * DIM_ + k] + xi * w_dt[h * 2 * DIM_ + DIM_ + k];
      ap[h]  += mag * w_pg[h * DIM_ + k];
    }
  }
  int b = bt / T_, t = bt % T_;
#pragma unroll
  for (int h = 0; h < H_; ++h) {
    float z  = adt[h] + b_dt[h] + dt_bias[h];
    float dt = (z > 20.f) ? z : log1pf(__expf(z));
    float pv = 1.f / (1.f + __expf(-(ap[h] + b_pg[h])));
    float g  = __expf(-dt) * (1.f - pv) + pv;
    size_t o = ((size_t)(b * H_ + h)) * T_ + t;
    p_out[o] = pv;
    g_out[o] = g;
  }
}

// ---------------------------------------------------------------------------
// QKV: complex GEMM (8192 x 3072 x 1024) via f32 WMMA, fused RoPE / scale / (1-p).
// Each wave: one 16-row M tile x 64-col N region (4 complex 16x16 subtiles).
// ---------------------------------------------------------------------------
__global__ __launch_bounds__(256) void qkv_kernel(
    const float* __restrict__ x, const float* __restrict__ wqr,
    const float* __restrict__ wqi, const float* __restrict__ p_in,
    float* __restrict__ qb, float* __restrict__ kb, float* __restrict__ vb) {
  const int lane = threadIdx.x & 31;
  const int hf   = lane >> 4;          // half-wave select (K pairing)
  const int ln   = lane & 15;
  int wid   = blockIdx.x * (blockDim.x >> 5) + (threadIdx.x >> 5);
  int mtile = wid / NREGQ;
  int nreg  = wid % NREGQ;
  int gm    = mtile * 16 + ln;         // global row = b*T + t
  const float* xrow = x + (size_t)gm * DIM_ * 2;

  v8f accR[4], accI[4];
#pragma unroll
  for (int s = 0; s < 4; ++s) { accR[s] = {}; accI[s] = {}; }

  for (int kk = 0; kk < DIM_; kk += 4) {
    int k0 = kk + 2 * hf;
    v4f av = *(const v4f*)(xrow + 2 * k0);       // {r0,i0,r1,i1}
    v2f ar = {av.x, av.z};
    v2f ai = {av.y, av.w};
    v2f nai = {-av.y, -av.w};
    __builtin_prefetch(xrow + 2 * k0 + 64, 0, 1);
#pragma unroll
    for (int s = 0; s < 4; ++s) {
      int n = nreg * 64 + s * 16 + ln;
      v2f br = *(const v2f*)(wqr + (size_t)n * DIM_ + k0);
      v2f bi = *(const v2f*)(wqi + (size_t)n * DIM_ + k0);
      accR[s] = wmma4(ar,  br, accR[s]);   // xr*wr
      accR[s] = wmma4(nai, bi, accR[s]);   // -xi*wi
      accI[s] = wmma4(ar,  bi, accI[s]);   // xr*wi
      accI[s] = wmma4(ai,  br, accI[s]);   // xi*wr
    }
  }

  const float LOG1E4_D = 9.210340371976184f / 64.f;
#pragma unroll
  for (int s = 0; s < 4; ++s) {
    int wbase = nreg * 64 + s * 16;
    int which = wbase / INNER_;               // 0=q 1=k 2=v
    int hidx  = (wbase % INNER_) / HD_;
    int hd    = (wbase % HD_) + ln;
    float freq = __expf(-(float)hd * LOG1E4_D);
    float* dst = (which == 0) ? qb : (which == 1) ? kb : vb;
#pragma unroll
    for (int r = 0; r < 8; ++r) {
      int M = r + 8 * hf;
      int g = mtile * 16 + M;
      int b = g >> 11, t = g & (T_ - 1);
      float er = accR[s][r], ei = accI[s][r];
      float orr, oii;
      if (which == 2) {
        float pv = p_in[((size_t)(b * H_ + hidx)) * T_ + t];
        float f  = 1.f - pv;
        orr = er * f; oii = ei * f;
      } else {
        float sn, cn;
        __sincosf((float)t * freq, &sn, &cn);
        float sc = (which == 0) ? 0.125f : 1.f;   // q scaled by HD^-0.5
        orr = (er * cn - ei * sn) * sc;
        oii = (er * sn + ei * cn) * sc;
      }
      size_t o = ((((size_t)(b * H_ + hidx)) * T_ + t) * HD_ + hd) * 2;
      v2f st = {orr, oii};
      *(v2f*)(dst + o) = st;
    }
  }
}

// ---------------------------------------------------------------------------
// Chunked retention scan: one block per (b,h). Interleaved complex q/k/v
// chunk planes are DMA'd global->LDS with async-to-LDS b128 (ASYNCcnt path),
// S kept as split-plane state, per-wave W-tile staging for A-operand reuse.
// ---------------------------------------------------------------------------
#define IRS 132                                 // interleaved row stride (floats), 16B aligned
#define PS  66                                  // S plane stride
#define SM_FLOATS (3 * CK * IRS + 2 * 64 * PS + 3 * CK + 8 * 2 * 16 * 17)

__global__ __launch_bounds__(256) void attn_kernel(
    const float* __restrict__ qg, const float* __restrict__ kg,
    const float* __restrict__ vg, const float* __restrict__ gam,
    float* __restrict__ yb, float* __restrict__ S_out) {
  extern __shared__ float sm[];
  float* qpl = sm;                              // CK x IRS, {r,i} interleaved
  float* kpl = qpl + CK * IRS;
  float* vpl = kpl + CK * IRS;
  float* Sr  = vpl + CK * IRS;                  // 64 x PS split planes
  float* Si  = Sr + 64 * PS;
  float* cs  = Si + 64 * PS;
  float* cm  = cs + CK;
  float* dl  = cm + CK;
  float* wst = dl + CK;                         // 8 waves * 2 * 16*17

  const int tid  = threadIdx.x;
  const int lane = tid & 31;
  const int wave = tid >> 5;
  const int hf   = lane >> 4;
  const int ln   = lane & 15;
  const int bh   = blockIdx.x;

  for (int e = tid; e < 64 * PS; e += 256) { Sr[e] = 0.f; Si[e] = 0.f; }
  __syncthreads();

  for (int c = 0; c < NCHUNK; ++c) {
    // ---- stage q/k/v chunk into padded interleaved planes ----
    size_t gbase = ((size_t)bh * T_ + (size_t)c * CK) * HD_ * 2;
    const float* gsrc0 = qg + gbase;
    const float* gsrc1 = kg + gbase;
    const float* gsrc2 = vg + gbase;
#pragma unroll
    for (int tn = 0; tn < 3; ++tn) {
      const float* gp = (tn == 0) ? gsrc0 : (tn == 1) ? gsrc1 : gsrc2;
      float* lp = (tn == 0) ? qpl : (tn == 1) ? kpl : vpl;
      for (int g = tid; g < CK * HD_ / 2; g += 256) {   // 16B granules
        int t = g >> 5, c2 = g & 31;
#ifdef ASYNC_COPY
        __builtin_amdgcn_global_load_async_to_lds_b128(
            (ai4*)(gp + 4 * g),
            (lds_ai4_p)(lp + t * IRS + 4 * c2),
            0, 0);
#else
        v4f tmp = *(const v4f*)(gp + 4 * g);
        *(v4f*)(lp + t * IRS + 4 * c2) = tmp;
#endif
      }
    }
#ifdef ASYNC_COPY
    asm volatile("s_wait_asynccnt 0x0" ::: "memory");
#endif
    if (tid == 0) {                             // serial 128-element scan
      const float* gpv = gam + (size_t)bh * T_ + (size_t)c * CK;
      float run = 0.f;
      for (int t = 0; t < CK; ++t) {
        run += -logf(gpv[t] + 1e-6f);
        cs[t] = run;
        cm[t] = __expf(-run);
      }
      float csl = run;
      for (int t = 0; t < CK; ++t) dl[t] = __expf(fminf(cs[t] - csl, 0.f));
    }
    __syncthreads();

    // ---- Y phase: wave owns row tile m = wave ----
    const int mt = wave;
    v8f accR[4], accI[4];
#pragma unroll
    for (int s = 0; s < 4; ++s) { accR[s] = {}; accI[s] = {}; }

    // Y2 = q @ S^T (complex), scaled by cum[t] afterwards
    for (int kk = 0; kk < HD_; kk += 4) {
      int e0 = kk + 2 * hf;
      int rq = mt * 16 + ln;
      v4f aq = *(const v4f*)(qpl + rq * IRS + 2 * e0);   // {r0,i0,r1,i1}
      v2f aqr = {aq.x, aq.z};
      v2f aqi = {aq.y, aq.w};
      v2f naqi = {-aq.y, -aq.w};
#pragma unroll
      for (int s = 0; s < 4; ++s) {
        int d = s * 16 + ln;
        v2f bsr = *(const v2f*)(Sr + d * PS + e0);
        v2f bsi = *(const v2f*)(Si + d * PS + e0);
        accR[s] = wmma4(aqr,  bsr, accR[s]);
        accR[s] = wmma4(naqi, bsi, accR[s]);
        accI[s] = wmma4(aqi,  bsr, accI[s]);
        accI[s] = wmma4(aqr,  bsi, accI[s]);
      }
    }
#pragma unroll
    for (int s = 0; s < 4; ++s)
#pragma unroll
      for (int r = 0; r < 8; ++r) {
        float f = cm[mt * 16 + r + 8 * hf];
        accR[s][r] *= f; accI[s][r] *= f;
      }

    // Y1: causal j-blocks. W tile -> decay -> LDS stage -> A-operand reuse.
    float* wsr = wst + wave * 2 * 16 * 17;
    float* wsi = wsr + 16 * 17;
    for (int j = 0; j <= mt; ++j) {
      v8f wR = {}, wI = {};
      for (int kk = 0; kk < HD_; kk += 4) {
        int e0 = kk + 2 * hf;
        int rq = mt * 16 + ln;
        int rk = j * 16 + ln;
        v4f aq = *(const v4f*)(qpl + rq * IRS + 2 * e0);
        v4f bk = *(const v4f*)(kpl + rk * IRS + 2 * e0);
        v2f aqr = {aq.x, aq.z};
        v2f aqi = {aq.y, aq.w};
        v2f naqr = {-aq.x, -aq.z};
        v2f bkr = {bk.x, bk.z};
        v2f bki = {bk.y, bk.w};
        wR = wmma4(aqr,  bkr, wR);    // qr*kr
        wR = wmma4(aqi,  bki, wR);    // qi*ki
        wI = wmma4(aqi,  bkr, wI);    // qi*kr
        wI = wmma4(naqr, bki, wI);    // -qr*ki
      }
#pragma unroll
      for (int r = 0; r < 8; ++r) {   // decay mask + stage (wave-private)
        int M = r + 8 * hf;
        int trow = mt * 16 + M, tcol = j * 16 + ln;
        float d = (tcol <= trow) ? __expf(fminf(cs[tcol] - cs[trow], 0.f)) : 0.f;
        wsr[M * 17 + ln] = wR[r] * d;
        wsi[M * 17 + ln] = wI[r] * d;
      }
      for (int kk = 0; kk < 16; kk += 4) {      // Y += A @ v
        int kb = kk + 2 * hf;
        v2f awr = {wsr[ln * 17 + kb], wsr[ln * 17 + kb + 1]};
        v2f awi = {wsi[ln * 17 + kb], wsi[ln * 17 + kb + 1]};
        v2f nawi = -awi;
        int t0 = j * 16 + kb;
#pragma unroll
        for (int s = 0; s < 4; ++s) {
          int d = s * 16 + ln;
          v2f e0v = *(const v2f*)(vpl + t0 * IRS + 2 * d);        // {vr,vi}@t0
          v2f e1v = *(const v2f*)(vpl + (t0 + 1) * IRS + 2 * d);  // {vr,vi}@t0+1
          v2f bvr = {e0v.x, e1v.x};
          v2f bvi = {e0v.y, e1v.y};
          accR[s] = wmma4(awr,  bvr, accR[s]);
          accR[s] = wmma4(nawi, bvi, accR[s]);
          accI[s] = wmma4(awr,  bvi, accI[s]);
          accI[s] = wmma4(awi,  bvr, accI[s]);
        }
      }
    }

    // store Y chunk
    float* yo = yb + (((size_t)bh * T_ + (size_t)c * CK) * HD_) * 2;
#pragma unroll
    for (int s = 0; s < 4; ++s)
#pragma unroll
      for (int r = 0; r < 8; ++r) {
        int t = mt * 16 + r + 8 * hf;
        int d = s * 16 + ln;
        v2f st = {accR[s][r], accI[s][r]};
        *(v2f*)(yo + ((size_t)t * HD_ + d) * 2) = st;
      }
    __syncthreads();   // all Y2 reads of old S complete

    // ---- S phase: S = S*total + (v'*D_last)^T @ k ; 2 tiles per wave ----
    float tot = cm[CK - 1];
    for (int tt = wave * 2; tt < wave * 2 + 2; ++tt) {
      int dvb = (tt >> 2) * 16, dkb = (tt & 3) * 16;
      v8f sR = {}, sI = {};
      for (int kk = 0; kk < CK; kk += 4) {
        int t0 = kk + 2 * hf;
        int dv = dvb + ln;
        float d0 = dl[t0], d1 = dl[t0 + 1];
        v2f vv0 = *(const v2f*)(vpl + t0 * IRS + 2 * dv);
        v2f vv1 = *(const v2f*)(vpl + (t0 + 1) * IRS + 2 * dv);
        v2f awr = {vv0.x * d0, vv1.x * d1};
        v2f awi = {vv0.y * d0, vv1.y * d1};
        v2f nawr = -awr;
        int dk = dkb + ln;
        v2f kk0 = *(const v2f*)(kpl + t0 * IRS + 2 * dk);
        v2f kk1 = *(const v2f*)(kpl + (t0 + 1) * IRS + 2 * dk);
        v2f bkr = {kk0.x, kk1.x};
        v2f bki = {kk0.y, kk1.y};
        sR = wmma4(awr,  bkr, sR);    // wv_r*kr
        sR = wmma4(awi,  bki, sR);    // wv_i*ki
        sI = wmma4(awi,  bkr, sI);    // wv_i*kr
        sI = wmma4(nawr, bki, sI);    // -wv_r*ki
      }
#pragma unroll
      for (int r = 0; r < 8; ++r) {
        int dv = dvb + r + 8 * hf, dk = dkb + ln;
        Sr[dv * PS + dk] = Sr[dv * PS + dk] * tot + sR[r];
        Si[dv * PS + dk] = Si[dv * PS + dk] * tot + sI[r];
      }
    }
    __syncthreads();
  }

  // final state -> d_out tail: (B,H,HD,HD,2)
  for (int e = tid; e < 64 * 64; e += 256) {
    int dv = e >> 6, dk = e & 63;
    size_t o = ((size_t)bh * HD_ * HD_ + e) * 2;
    S_out[o]     = Sr[dv * PS + dk];
    S_out[o + 1] = Si[dv * PS + dk];
  }
}

// ---------------------------------------------------------------------------
// Output projection: complex GEMM (8192 x 1024 x 1024), A gathered from
// head-major y buffer.
// ---------------------------------------------------------------------------
__global__ __launch_bounds__(256) void oproj_kernel(
    const float* __restrict__ yb, const float* __restrict__ wor,
    const float* __restrict__ woi, float* __restrict__ out) {
  const int lane = threadIdx.x & 31;
  const int hf   = lane >> 4;
  const int ln   = lane & 15;
  int wid   = blockIdx.x * (blockDim.x >> 5) + (threadIdx.x >> 5);
  int mtile = wid / NREGO;
  int nreg  = wid % NREGO;
  int gm = mtile * 16 + ln;
  int b = gm >> 11, t = gm & (T_ - 1);

  v8f accR[4], accI[4];
#pragma unroll
  for (int s = 0; s < 4; ++s) { accR[s] = {}; accI[s] = {}; }

  for (int kk = 0; kk < INNER_; kk += 4) {
    int k0 = kk + 2 * hf;
    int h = k0 >> 6, hd = k0 & 63;
    const float* yp = yb + ((((size_t)(b * H_ + h)) * T_ + t) * HD_ + hd) * 2;
    v4f av = *(const v4f*)yp;
    v2f ar = {av.x, av.z};
    v2f ai = {av.y, av.w};
    v2f nai = {-av.y, -av.w};
#pragma unroll
    for (int s = 0; s < 4; ++s) {
      int n = nreg * 64 + s * 16 + ln;
      v2f br = *(const v2f*)(wor + (size_t)n * INNER_ + k0);
      v2f bi = *(const v2f*)(woi + (size_t)n * INNER_ + k0);
      __builtin_prefetch(wor + (size_t)n * INNER_ + k0 + 64, 0, 1);
      accR[s] = wmma4(ar,  br, accR[s]);
      accR[s] = wmma4(nai, bi, accR[s]);
      accI[s] = wmma4(ar,  bi, accI[s]);
      accI[s] = wmma4(ai,  br, accI[s]);
    }
  }
#pragma unroll
  for (int s = 0; s < 4; ++s)
#pragma unroll
    for (int r = 0; r < 8; ++r) {
      int g = mtile * 16 + r + 8 * hf;
      int n = nreg * 64 + s * 16 + ln;
      v2f st = {accR[s][r], accI[s][r]};
      *(v2f*)(out + ((size_t)g * DIM_ + n) * 2) = st;
    }
}

// ---------------------------------------------------------------------------
extern "C" void kernel_launch(void* const* d_in, const int* in_sizes, int n_in,
                              void* d_out, int out_size, void* d_ws,
                              size_t ws_size, hipStream_t stream) {
  const float* x       = (const float*)d_in[0];
  const float* w_qkv_r = (const float*)d_in[1];
  const float* w_qkv_i = (const float*)d_in[2];
  const float* w_o_r   = (const float*)d_in[3];
  const float* w_o_i   = (const float*)d_in[4];
  const float* w_dt    = (const float*)d_in[5];
  const float* b_dt    = (const float*)d_in[6];
  const float* dt_bias = (const float*)d_in[7];
  const float* w_pg    = (const float*)d_in[8];
  const float* b_pg    = (const float*)d_in[9];

  float* ws = (float*)d_ws;
  size_t off = 0;
  float* p_buf = ws + off; off += (size_t)B_ * H_ * T_;
  float* g_buf = ws + off; off += (size_t)B_ * H_ * T_;
  float* q_buf = ws + off; off += (size_t)B_ * H_ * T_ * HD_ * 2;
  float* k_buf = ws + off; off += (size_t)B_ * H_ * T_ * HD_ * 2;
  float* v_buf = ws + off; off += (size_t)B_ * H_ * T_ * HD_ * 2;
  float* y_buf = ws + off; off += (size_t)B_ * H_ * T_ * HD_ * 2;

  float* out_main = (float*)d_out;
  float* S_out    = out_main + (size_t)B_ * T_ * DIM_ * 2;

  // 1) gates
  gate_kernel<<<(B_ * T_) / 256, 256, 0, stream>>>(x, w_dt, b_dt, dt_bias,
                                                   w_pg, b_pg, p_buf, g_buf);
  // 2) qkv projection + RoPE + gating, (512 M-tiles)*(48 N-regions) waves
  qkv_kernel<<<(512 * NREGQ) / 8, 256, 0, stream>>>(x, w_qkv_r, w_qkv_i, p_buf,
                                                    q_buf, k_buf, v_buf);
  // 3) chunked retention scan, one block per (b,h)
  const size_t smem = (size_t)SM_FLOATS * sizeof(float);
  (void)hipFuncSetAttribute((const void*)attn_kernel,
                            hipFuncAttributeMaxDynamicSharedMemorySize,
                            (int)smem);
  attn_kernel<<<B_ * H_, 256, smem, stream>>>(q_buf, k_buf, v_buf, g_buf,
                                              y_buf, S_out);
  // 4) output projection, (512 M-tiles)*(16 N-regions) waves
  oproj_kernel<<<(512 * NREGO) / 8, 256, 0, stream>>>(y_buf, w_o_r, w_o_i,
                                                      out_main);
}